// VIN_18202071400728
// MI455X (gfx1250) — compile-verified
//
#include <hip/hip_runtime.h>
#include <hip/hip_bf16.h>

typedef __attribute__((ext_vector_type(2))) float v2f;
typedef __attribute__((ext_vector_type(8))) float v8f;

#define IMS 64
#define PADW 66
#define PIMG (PADW * PADW) /* 4356 */

__global__ __launch_bounds__(256) void vin_fused_kernel(
    const float* __restrict__ X,
    const float* __restrict__ h_w,
    const float* __restrict__ h_b,
    const float* __restrict__ r_w,
    const float* __restrict__ q_w,
    const float* __restrict__ w,
    const float* __restrict__ fc_w,
    const int* __restrict__ S1,
    const int* __restrict__ S2,
    const int* __restrict__ kptr,
    float* __restrict__ out)
{
    // rv[0..PIMG)      : r, zero-padded 66x66 (interior at (i+1,j+1))
    // rv[PIMG..2*PIMG) : v, zero-padded 66x66
    __shared__ float rv[2 * PIMG];
    __shared__ float vtmp[IMS * IMS];
    __shared__ float weff[20];       // [0..17] fused 2ch 3x3 weights, [18] fused bias
    __shared__ float wcat[10 * 18];  // w_cat[o][c*9 + ky*3 + kx]
    __shared__ float qsh[10];
    __shared__ float lsh[8];
    __shared__ float esh[8];

    const int b = blockIdx.x;
    const int tid = threadIdx.x;

    // ---- Stage 0: fuse the linear hidden layer: Weff = sum_c r_w[c]*h_w[c], beff = sum_c r_w[c]*h_b[c]
    if (tid < 19) {
        float acc = 0.f;
        for (int c = 0; c < 150; ++c) {
            float rc = r_w[c];
            acc = fmaf(rc, (tid < 18) ? h_w[c * 18 + tid] : h_b[c], acc);
        }
        weff[tid] = acc;
    }
    // w_cat = concat([q_w, w], axis=1): channel 0 weights = q_w, channel 1 = w
    if (tid >= 64 && tid < 244) {
        int kidx = tid - 64;
        int o = kidx / 18, p = kidx % 18;
        wcat[kidx] = (p < 9) ? q_w[o * 9 + p] : w[o * 9 + (p - 9)];
    }
    for (int i = tid; i < 2 * PIMG; i += 256) rv[i] = 0.f;  // zero halos; v starts at 0
    __syncthreads();

    // ---- Stage 1: r = conv(X, Weff, SAME) + beff  (into padded LDS)
    const float* Xb = X + (size_t)b * 2 * IMS * IMS;
    for (int p = tid; p < IMS * IMS; p += 256) {
        int i = p >> 6, j = p & 63;
        float acc = weff[18];
#pragma unroll
        for (int c = 0; c < 2; ++c)
#pragma unroll
            for (int ky = 0; ky < 3; ++ky)
#pragma unroll
                for (int kx = 0; kx < 3; ++kx) {
                    int ii = i + ky - 1, jj = j + kx - 1;
                    float xv = (ii >= 0 && ii < IMS && jj >= 0 && jj < IMS)
                                   ? Xb[c * IMS * IMS + ii * IMS + jj]
                                   : 0.f;
                    acc = fmaf(weff[c * 9 + ky * 3 + kx], xv, acc);
                }
        rv[(i + 1) * PADW + (j + 1)] = acc;
    }
    __syncthreads();

    const int lane = tid & 31;
    const int half = lane >> 4;  // 0: K rows {0,1} / M 0..7 of D; 1: K rows {2,3} / M 8..15
    const int n = lane & 15;     // A: row M; B/D: column N (pixel within tile)
    const int wave = tid >> 5;   // 8 waves

    // ---- Preload A operand (16x4 f32 layout): lane n holds W[n][k0], W[n][k0+1]; k0 = 4c + 2*half
    // K rows 18,19 and M rows 10..15 are zero-padded weights.
    v2f A[5];
#pragma unroll
    for (int c = 0; c < 5; ++c) {
        int k0 = c * 4 + half * 2;
        A[c].x = (n < 10 && k0 < 18) ? wcat[n * 18 + k0] : 0.f;
        A[c].y = (n < 10 && k0 + 1 < 18) ? wcat[n * 18 + k0 + 1] : 0.f;
    }

    // ---- Precompute per-lane B gather offsets (K -> {r|v} + (ky,kx) offset).
    // For pad rows K=18,19 the A weights above are zero, so the B value is
    // irrelevant (any finite value x 0 = 0): clamp to k=0, a valid r-patch
    // address, and load unconditionally -> no exec-masked loads in the hot loop.
    int boff[5][2];
#pragma unroll
    for (int c = 0; c < 5; ++c) {
#pragma unroll
        for (int e = 0; e < 2; ++e) {
            int k = c * 4 + half * 2 + e;
            int kc = (k < 18) ? k : 0;
            int src = (kc >= 9) ? 1 : 0;
            int p = kc - src * 9;
            int ky = p / 3, kx = p - ky * 3;
            boff[c][e] = src * PIMG + ky * PADW + kx;
        }
    }

    // ---- Stage 2: k value-iteration steps: q = W x im2col([r,v]); v = max_o q
    const int kk = kptr[0];
    for (int it = 0; it < kk; ++it) {
        for (int t = wave; t < 256; t += 8) {  // uniform per wave -> EXEC all-ones at WMMA
            int row = t >> 2;                  // 0..63
            int col0 = (t & 3) << 4;           // 0,16,32,48
            int base = row * PADW + col0 + n;  // patch(ky,kx) @ rv[base + ky*66 + kx]
            v8f acc = {0.f, 0.f, 0.f, 0.f, 0.f, 0.f, 0.f, 0.f};
#pragma unroll
            for (int c = 0; c < 5; ++c) {
                v2f bv;
                bv.x = rv[base + boff[c][0]];
                bv.y = rv[base + boff[c][1]];
                acc = __builtin_amdgcn_wmma_f32_16x16x4_f32(
                    false, A[c], false, bv, (short)0, acc, false, false);
            }
            // D layout: VGPR d = channel d (lanes 0-15) / channel d+8 (lanes 16-31)
            float mlow = fmaxf(fmaxf(fmaxf(acc[0], acc[1]), fmaxf(acc[2], acc[3])),
                               fmaxf(fmaxf(acc[4], acc[5]), fmaxf(acc[6], acc[7])));
            float mhigh = fmaxf(acc[0], acc[1]);  // channels 8,9
            float mx = half ? mhigh : mlow;
            float ov = __shfl_xor(mx, 16, 32);
            float vmax = fmaxf(mx, ov);
            if (half == 0) vtmp[row * IMS + col0 + n] = vmax;
        }
        __syncthreads();
        for (int p = tid; p < IMS * IMS; p += 256)
            rv[PIMG + ((p >> 6) + 1) * PADW + (p & 63) + 1] = vtmp[p];
        __syncthreads();
    }

    // ---- Stage 3: final conv at the single sampled pixel, FC, softmax
    const int s1 = S1[b], s2 = S2[b];
    if (tid < 10) {
        float acc = 0.f;
        int baseq = s1 * PADW + s2;
#pragma unroll
        for (int k = 0; k < 18; ++k) {
            int src = (k >= 9) ? 1 : 0;
            int p = k - src * 9;
            int ky = p / 3, kx = p - ky * 3;
            acc = fmaf(wcat[tid * 18 + k], rv[src * PIMG + baseq + ky * PADW + kx], acc);
        }
        qsh[tid] = acc;
    }
    __syncthreads();
    if (tid < 8) {
        float acc = 0.f;
#pragma unroll
        for (int o = 0; o < 10; ++o) acc = fmaf(fc_w[tid * 10 + o], qsh[o], acc);
        lsh[tid] = acc;
        out[b * 8 + tid] = acc;  // logits
    }
    __syncthreads();
    if (tid < 8) {
        float mx = lsh[0];
#pragma unroll
        for (int j = 1; j < 8; ++j) mx = fmaxf(mx, lsh[j]);
        esh[tid] = __expf(lsh[tid] - mx);
    }
    __syncthreads();
    if (tid < 8) {
        float s = 0.f;
#pragma unroll
        for (int j = 0; j < 8; ++j) s += esh[j];
        out[128 * 8 + b * 8 + tid] = esh[tid] / s;  // probs
    }
}

extern "C" void kernel_launch(void* const* d_in, const int* in_sizes, int n_in,
                              void* d_out, int out_size, void* d_ws, size_t ws_size,
                              hipStream_t stream) {
    const float* X    = (const float*)d_in[0];
    const float* h_w  = (const float*)d_in[1];
    const float* h_b  = (const float*)d_in[2];
    const float* r_w  = (const float*)d_in[3];
    const float* q_w  = (const float*)d_in[4];
    const float* w    = (const float*)d_in[5];
    const float* fc_w = (const float*)d_in[6];
    const int* S1     = (const int*)d_in[7];
    const int* S2     = (const int*)d_in[8];
    // d_in[9], d_in[10] = O1, O2 (unused by reference)
    const int* kptr   = (const int*)d_in[11];
    float* out = (float*)d_out;

    vin_fused_kernel<<<128, 256, 0, stream>>>(X, h_w, h_b, r_w, q_w, w, fc_w,
                                              S1, S2, kptr, out);
}